// ResGCNBlock_60498909331788
// MI455X (gfx1250) — compile-verified
//
#include <hip/hip_runtime.h>

typedef __attribute__((ext_vector_type(2))) float v2f;
typedef __attribute__((ext_vector_type(8))) float v8f;

#define DIM 128
#define BN_EPS 1e-5f

// ---------------- init: deg = 1.0 (self-loop), zero BN partial sums ----------------
__global__ void rgcn_init(float* __restrict__ deg, float* __restrict__ sums, int n) {
    int i = blockIdx.x * blockDim.x + threadIdx.x;
    if (i < n) deg[i] = 1.0f;
    if (i < 2 * DIM) sums[i] = 0.0f;
}

// ---------------- degree over dst ----------------
__global__ void rgcn_degree(const int* __restrict__ dst, float* __restrict__ deg, int nE) {
    int e = blockIdx.x * blockDim.x + threadIdx.x;
    if (e < nE) atomicAdd(&deg[dst[e]], 1.0f);
}

// ---------------- h = x @ W via V_WMMA_F32_16X16X4_F32 ----------------
// One wave32 per 16x16 output tile. K=128 -> 32 chained WMMAs.
// A (16x4 f32): lanes 0-15 hold M=0..15 {K=0,K=1}; lanes 16-31 hold {K=2,K=3}.
// B (4x16 f32): lanes 0-15 hold N=0..15 rows {K=0,K=1}; lanes 16-31 {K=2,K=3}.
// C/D (16x16 f32): VGPR r = M=r (lanes 0-15) / M=r+8 (lanes 16-31), N = lane%16.
__global__ void rgcn_gemm(const float* __restrict__ x, const float* __restrict__ W,
                          float* __restrict__ h, int n) {
    int wave = (int)((blockIdx.x * blockDim.x + threadIdx.x) >> 5);
    int lane = threadIdx.x & 31;
    const int nTilesN = DIM / 16;               // 8
    int mTile = wave / nTilesN;
    int nTile = wave - mTile * nTilesN;
    int m0 = mTile * 16, n0 = nTile * 16;
    if (m0 >= n) return;

    int row  = lane & 15;
    int half = lane >> 4;                       // 0: K=0,1   1: K=2,3
    const float* xr = x + (size_t)(m0 + row) * DIM;

    v8f acc = {};
    #pragma unroll 4
    for (int k = 0; k < DIM; k += 4) {
        int kb = k + half * 2;
        v2f a = { xr[kb], xr[kb + 1] };
        v2f b = { W[(size_t)kb * DIM + n0 + row],
                  W[(size_t)(kb + 1) * DIM + n0 + row] };
        acc = __builtin_amdgcn_wmma_f32_16x16x4_f32(
                  false, a, false, b, (short)0, acc, false, false);
    }
    #pragma unroll
    for (int r = 0; r < 8; ++r)
        h[(size_t)(m0 + r + half * 8) * DIM + n0 + row] = acc[r];
}

// ---------------- agg = h * dinv^2 + b (self-loop term + bias); store dinv ----------------
__global__ void rgcn_agg_init(const float* __restrict__ h, const float* __restrict__ deg,
                              const float* __restrict__ b, float* __restrict__ agg,
                              float* __restrict__ dinv, int n) {
    int idx = blockIdx.x * blockDim.x + threadIdx.x;     // over n*DIM
    if (idx >= n * DIM) return;
    int i = idx >> 7;
    int d = idx & (DIM - 1);
    float di = rsqrtf(fmaxf(deg[i], 1e-12f));
    if (d == 0) dinv[i] = di;
    agg[idx] = h[idx] * di * di + b[d];
}

// ---------------- edge scatter: one wave32 per edge, 4 dims per lane ----------------
__global__ void rgcn_scatter(const float* __restrict__ h, const int* __restrict__ src,
                             const int* __restrict__ dst, const float* __restrict__ dinv,
                             float* __restrict__ agg, int nE) {
    int wave = (int)((blockIdx.x * blockDim.x + threadIdx.x) >> 5);
    int lane = threadIdx.x & 31;
    if (wave >= nE) return;
    int s = src[wave];
    int d = dst[wave];
    float norm = dinv[s] * dinv[d];
    const float4* hs = (const float4*)(h + (size_t)s * DIM);
    float4 v = hs[lane];                                  // coalesced 512B/wave gather
    float* ad = agg + (size_t)d * DIM + lane * 4;
    atomicAdd(ad + 0, v.x * norm);                        // global_atomic_add_f32, L2-resident
    atomicAdd(ad + 1, v.y * norm);
    atomicAdd(ad + 2, v.z * norm);
    atomicAdd(ad + 3, v.w * norm);
}

// ---------------- BN stats: per-feature sum & sumsq ----------------
__global__ void rgcn_bn_stats(const float* __restrict__ agg, float* __restrict__ sums, int n) {
    int d      = threadIdx.x & (DIM - 1);
    int rowOff = threadIdx.x >> 7;                        // 0 or 1 (256 threads = 2 rows/iter)
    float s = 0.0f, sq = 0.0f;
    for (int r = blockIdx.x * 2 + rowOff; r < n; r += gridDim.x * 2) {
        float v = agg[(size_t)r * DIM + d];               // coalesced
        s += v; sq += v * v;
    }
    __shared__ float ls[256], lq[256];
    ls[threadIdx.x] = s; lq[threadIdx.x] = sq;
    __syncthreads();
    if (threadIdx.x < DIM) {
        atomicAdd(&sums[d],       ls[d] + ls[d + DIM]);
        atomicAdd(&sums[DIM + d], lq[d] + lq[d + DIM]);
    }
}

// ---------------- epilogue: BN normalize + relu + residual ----------------
__global__ void rgcn_finalize(const float* __restrict__ agg, const float* __restrict__ x,
                              const float* __restrict__ sums, const float* __restrict__ gamma,
                              const float* __restrict__ beta, float* __restrict__ out, int n) {
    int idx = blockIdx.x * blockDim.x + threadIdx.x;      // over n*DIM
    if (idx >= n * DIM) return;
    int d = idx & (DIM - 1);
    float invN = 1.0f / (float)n;
    float mean = sums[d] * invN;
    float var  = sums[DIM + d] * invN - mean * mean;
    float inv  = rsqrtf(var + BN_EPS);
    float y = gamma[d] * (agg[idx] - mean) * inv + beta[d];
    out[idx] = fmaxf(y, 0.0f) + x[idx];
}

extern "C" void kernel_launch(void* const* d_in, const int* in_sizes, int n_in,
                              void* d_out, int out_size, void* d_ws, size_t ws_size,
                              hipStream_t stream) {
    const float* x     = (const float*)d_in[0];
    const float* W     = (const float*)d_in[1];
    const float* b     = (const float*)d_in[2];
    const float* gamma = (const float*)d_in[3];
    const float* beta  = (const float*)d_in[4];
    const int*   eidx  = (const int*)d_in[5];

    const int n  = in_sizes[0] / DIM;        // 50000
    const int nE = in_sizes[5] / 2;          // 1600000
    const int* src = eidx;
    const int* dst = eidx + nE;

    // workspace layout
    float* h    = (float*)d_ws;              // n*DIM
    float* agg  = h   + (size_t)n * DIM;     // n*DIM
    float* deg  = agg + (size_t)n * DIM;     // n
    float* dinv = deg + n;                   // n
    float* sums = dinv + n;                  // 2*DIM

    const int elems = n * DIM;

    // 1) init deg=1 (self-loops), zero BN sums
    rgcn_init<<<(n + 255) / 256, 256, 0, stream>>>(deg, sums, n);
    // 2) degree over dst
    rgcn_degree<<<(nE + 255) / 256, 256, 0, stream>>>(dst, deg, nE);
    // 3) h = x @ W  (WMMA f32, one wave per 16x16 tile; 8 waves/block)
    {
        int tiles  = (n / 16) * (DIM / 16);  // 3125 * 8 = 25000
        int blocks = (tiles + 7) / 8;
        rgcn_gemm<<<blocks, 256, 0, stream>>>(x, W, h, n);
    }
    // 4) agg = h*dinv^2 + b ; store dinv
    rgcn_agg_init<<<(elems + 255) / 256, 256, 0, stream>>>(h, deg, b, agg, dinv, n);
    // 5) edge scatter (one wave32 per edge, 8 edges/block)
    rgcn_scatter<<<(nE + 7) / 8, 256, 0, stream>>>(h, src, dst, dinv, agg, nE);
    // 6) BN stats
    rgcn_bn_stats<<<512, 256, 0, stream>>>(agg, sums, n);
    // 7) normalize + relu + residual
    rgcn_finalize<<<(elems + 255) / 256, 256, 0, stream>>>(agg, x, sums, gamma, beta, (float*)d_out, n);
}